// GCNet_55594056679487
// MI455X (gfx1250) — compile-verified
//
#include <hip/hip_runtime.h>
#include <hip/hip_bf16.h>

typedef __attribute__((ext_vector_type(2))) float v2f;
typedef __attribute__((ext_vector_type(8))) float v8f;

#define HID 128

// ---------------------------------------------------------------------------
// zero fill
// ---------------------------------------------------------------------------
__global__ void zero_kernel(float* __restrict__ p, long long n) {
    long long i = (long long)blockIdx.x * blockDim.x + threadIdx.x;
    long long stride = (long long)gridDim.x * blockDim.x;
    for (; i < n; i += stride) p[i] = 0.0f;
}

// ---------------------------------------------------------------------------
// degree count: deg[dst[e]] += 1
// ---------------------------------------------------------------------------
__global__ void deg_kernel(const long long* __restrict__ dst, float* __restrict__ deg,
                           long long n_edges) {
    long long e = (long long)blockIdx.x * blockDim.x + threadIdx.x;
    if (e >= n_edges) return;
    unsafeAtomicAdd(&deg[dst[e]], 1.0f);
}

// deg -> dinv = rsqrt(deg + 2)  (in place)
__global__ void dinv_kernel(float* __restrict__ d, int n) {
    int i = blockIdx.x * blockDim.x + threadIdx.x;
    if (i >= n) return;
    d[i] = rsqrtf(d[i] + 2.0f);
}

// ---------------------------------------------------------------------------
// WMMA GEMM: Y[n_rows x 128] = X[n_rows x 128] @ W[128 x 128]
// epilogue: 0 = raw, 1 = tanh(acc + bias)
// Block = 256 threads (8 waves). W is staged once into LDS (64 KB), then the
// block grid-strides over 16-row tiles; wave w owns column tile w. Inner loop
// per 16x16x4 wmma: 1x global_load_b64 (A) + 2x ds_load_b32 (B) + wmma.
// Row index is CLAMPED (not predicated) so EXEC stays all-ones for WMMA and
// no per-iteration exec save/restore is emitted; OOB rows are masked at store.
// ---------------------------------------------------------------------------
__global__ __launch_bounds__(256)
void gemm128_kernel(const float* __restrict__ X, const float* __restrict__ W,
                    const float* __restrict__ bias, float* __restrict__ Y,
                    int n_rows, int n_tiles, int epilogue) {
    __shared__ float lds_w[HID * HID];          // 64 KB
    {
        const float4* ws = (const float4*)W;
        float4*       ld = (float4*)lds_w;
        for (int i = threadIdx.x; i < (HID * HID) / 4; i += blockDim.x)
            ld[i] = ws[i];
    }
    __syncthreads();

    const int wave = threadIdx.x >> 5;          // 0..7 -> column tile
    const int lane = threadIdx.x & 31;
    const int hl   = lane & 15;                 // 0..15
    const int hi   = lane >> 4;                 // 0 or 1
    const int col  = wave * 16 + hl;

    for (int tile = blockIdx.x; tile < n_tiles; tile += gridDim.x) {
        const int row_base = tile * 16;
        int arow = row_base + hl;
        if (arow > n_rows - 1) arow = n_rows - 1;      // clamp, no divergence
        const float* xrow = X + (size_t)arow * HID + hi * 2;

        v8f acc = {};
        #pragma unroll
        for (int k0 = 0; k0 < HID; k0 += 4) {
            // A frag: lanes 0-15 -> (K=k0,k0+1), lanes 16-31 -> (K=k0+2,k0+3)
            v2f a;
            a.x = xrow[k0];
            a.y = xrow[k0 + 1];
            // B frag from LDS, same K split
            const int ka = k0 + hi * 2;
            v2f b;
            b.x = lds_w[ka * HID + col];
            b.y = lds_w[(ka + 1) * HID + col];
            acc = __builtin_amdgcn_wmma_f32_16x16x4_f32(
                /*neg_a=*/false, a, /*neg_b=*/false, b,
                /*c_mod=*/(short)0, acc, /*reuse_a=*/false, /*reuse_b=*/false);
        }

        // C/D layout: VGPR r -> (M = r + 8*hi, N = hl)
        #pragma unroll
        for (int r = 0; r < 8; ++r) {
            int row = row_base + r + 8 * hi;
            if (row < n_rows) {
                float v = acc[r];
                if (epilogue == 1) v = tanhf(v + bias[col]);
                Y[(size_t)row * HID + col] = v;
            }
        }
    }
}

// ---------------------------------------------------------------------------
// edge scatter: AGG[dst] += dinv[src]*dinv[dst] * XW[src]
// one wave per edge, one float4 (4 channels) per lane
// ---------------------------------------------------------------------------
__global__ __launch_bounds__(256)
void scatter_kernel(const float* __restrict__ XW, const long long* __restrict__ src,
                    const long long* __restrict__ dst, const float* __restrict__ dinv,
                    float* __restrict__ AGG, long long n_edges) {
    const int wpb = blockDim.x >> 5;
    long long e = (long long)blockIdx.x * wpb + (threadIdx.x >> 5);
    if (e >= n_edges) return;
    const int lane = threadIdx.x & 31;
    const long long s = src[e];
    const long long d = dst[e];
    const float coeff = dinv[s] * dinv[d];
    const float4* xr = (const float4*)(XW + (size_t)s * HID);
    float* ar = AGG + (size_t)d * HID;
    float4 v = xr[lane];
    unsafeAtomicAdd(&ar[lane * 4 + 0], v.x * coeff);
    unsafeAtomicAdd(&ar[lane * 4 + 1], v.y * coeff);
    unsafeAtomicAdd(&ar[lane * 4 + 2], v.z * coeff);
    unsafeAtomicAdd(&ar[lane * 4 + 3], v.w * coeff);
}

// ---------------------------------------------------------------------------
// finalize GCN layer: H = tanh(AGG + XW * 2*dinv^2 + bias)   (in place OK)
// ---------------------------------------------------------------------------
__global__ void finalize_kernel(const float* __restrict__ AGG, const float* __restrict__ XW,
                                const float* __restrict__ dinv, const float* __restrict__ bias,
                                float* __restrict__ H, int n_rows) {
    long long idx = (long long)blockIdx.x * blockDim.x + threadIdx.x;
    long long total = (long long)n_rows * HID;
    if (idx >= total) return;
    int n = (int)(idx >> 7);      // /128
    int c = (int)(idx & 127);
    float di = dinv[n];
    float v = AGG[idx] + XW[idx] * (2.0f * di * di) + bias[c];
    H[idx] = tanhf(v);
}

// ---------------------------------------------------------------------------
// output head: OUT[n, 0..2] = H[n, :] @ lw4[128x3] + lb4
// one wave per node, shuffle reduction
// ---------------------------------------------------------------------------
__global__ __launch_bounds__(256)
void out_kernel(const float* __restrict__ H, const float* __restrict__ W,
                const float* __restrict__ b, float* __restrict__ OUT, int n) {
    const int wpb = blockDim.x >> 5;
    int node = blockIdx.x * wpb + (threadIdx.x >> 5);
    if (node >= n) return;
    const int lane = threadIdx.x & 31;
    float s0 = 0.0f, s1 = 0.0f, s2 = 0.0f;
    const float* hr = H + (size_t)node * HID;
    #pragma unroll
    for (int k = lane; k < HID; k += 32) {
        float h = hr[k];
        s0 += h * W[k * 3 + 0];
        s1 += h * W[k * 3 + 1];
        s2 += h * W[k * 3 + 2];
    }
    #pragma unroll
    for (int off = 16; off > 0; off >>= 1) {
        s0 += __shfl_xor(s0, off, 32);
        s1 += __shfl_xor(s1, off, 32);
        s2 += __shfl_xor(s2, off, 32);
    }
    if (lane == 0) {
        OUT[(size_t)node * 3 + 0] = s0 + b[0];
        OUT[(size_t)node * 3 + 1] = s1 + b[1];
        OUT[(size_t)node * 3 + 2] = s2 + b[2];
    }
}

// ---------------------------------------------------------------------------
extern "C" void kernel_launch(void* const* d_in, const int* in_sizes, int n_in,
                              void* d_out, int out_size, void* d_ws, size_t ws_size,
                              hipStream_t stream) {
    const float*     x    = (const float*)d_in[0];
    const long long* eidx = (const long long*)d_in[1];     // int64 [2, E]
    const float* W1  = (const float*)d_in[2];
    const float* b1  = (const float*)d_in[3];
    const float* W2  = (const float*)d_in[4];
    const float* b2  = (const float*)d_in[5];
    const float* lw1 = (const float*)d_in[6];
    const float* lb1 = (const float*)d_in[7];
    const float* lw2 = (const float*)d_in[8];
    const float* lb2 = (const float*)d_in[9];
    const float* lw3 = (const float*)d_in[10];
    const float* lb3 = (const float*)d_in[11];
    const float* lw4 = (const float*)d_in[12];
    const float* lb4 = (const float*)d_in[13];
    float* out = (float*)d_out;

    const int       N = in_sizes[0] / HID;
    const long long E = (long long)in_sizes[1] / 2;
    const long long* src = eidx;
    const long long* dst = eidx + E;

    const long long NH = (long long)N * HID;
    float* bufA = (float*)d_ws;            // [N, 128]
    float* bufB = bufA + NH;               // [N, 128]
    float* dinv = bufB + NH;               // [N]   (deg -> dinv in place)

    const int T = 256;
    const int n_tiles    = (N + 15) / 16;
    const int gemm_grid  = n_tiles < 2048 ? n_tiles : 2048;   // persistent blocks
    const int elem_grid  = (int)((NH + T - 1) / T);
    const int node_grid  = (N + T - 1) / T;
    const int edgeT_grid = (int)((E + T - 1) / T);
    const int waves_pb   = T / 32;
    const int edgeW_grid = (int)((E + waves_pb - 1) / waves_pb);
    const int oc_grid    = (N + waves_pb - 1) / waves_pb;

    // degrees -> dinv
    zero_kernel<<<1024, T, 0, stream>>>(dinv, N);
    deg_kernel<<<edgeT_grid, T, 0, stream>>>(dst, dinv, E);
    dinv_kernel<<<node_grid, T, 0, stream>>>(dinv, N);

    // ---- GCN layer 1 ----
    gemm128_kernel<<<gemm_grid, T, 0, stream>>>(x, W1, nullptr, bufA, N, n_tiles, 0);   // xw1
    zero_kernel<<<4096, T, 0, stream>>>(bufB, NH);
    scatter_kernel<<<edgeW_grid, T, 0, stream>>>(bufA, src, dst, dinv, bufB, E);
    finalize_kernel<<<elem_grid, T, 0, stream>>>(bufB, bufA, dinv, b1, bufB, N);        // h1 in bufB

    // ---- GCN layer 2 ----
    gemm128_kernel<<<gemm_grid, T, 0, stream>>>(bufB, W2, nullptr, bufA, N, n_tiles, 0); // xw2
    zero_kernel<<<4096, T, 0, stream>>>(bufB, NH);
    scatter_kernel<<<edgeW_grid, T, 0, stream>>>(bufA, src, dst, dinv, bufB, E);
    finalize_kernel<<<elem_grid, T, 0, stream>>>(bufB, bufA, dinv, b2, bufB, N);        // h2 in bufB

    // ---- MLP (fused bias + tanh) ----
    gemm128_kernel<<<gemm_grid, T, 0, stream>>>(bufB, lw1, lb1, bufA, N, n_tiles, 1);
    gemm128_kernel<<<gemm_grid, T, 0, stream>>>(bufA, lw2, lb2, bufB, N, n_tiles, 1);
    gemm128_kernel<<<gemm_grid, T, 0, stream>>>(bufB, lw3, lb3, bufA, N, n_tiles, 1);

    // ---- head 128 -> 3 ----
    out_kernel<<<oc_grid, T, 0, stream>>>(bufA, lw4, lb4, out, N);
}